// VectorQuantize_75642964017479
// MI455X (gfx1250) — compile-verified
//
#include <hip/hip_runtime.h>

typedef __attribute__((ext_vector_type(2))) float v2f;
typedef __attribute__((ext_vector_type(8))) float v8f;

#define NTOK   65536   // B*T = 16*4096
#define IN_DIM 512
#define DCODE  8
#define KCODES 1024
#define AUGW   12      // augmented codebook row: [-2c(8) | ||c||^2 | pad(3)]

// --------- kernel 0: pad w_in to 16 rows, augment codebook, zero losses ---------
__global__ __launch_bounds__(256)
void vq_prep(const float* __restrict__ codebook,
             const float* __restrict__ w_in,
             float* __restrict__ cb_aug,
             float* __restrict__ w_in_pad,   // [16][512], rows 8..15 zero
             float* __restrict__ loss_slots) {
  const int t = blockIdx.x * blockDim.x + threadIdx.x;   // 8192 threads
  if (t < KCODES) {
    float n2 = 0.f;
    float r[DCODE];
#pragma unroll
    for (int d = 0; d < DCODE; ++d) { r[d] = codebook[t * DCODE + d]; n2 += r[d] * r[d]; }
#pragma unroll
    for (int d = 0; d < DCODE; ++d) cb_aug[t * AUGW + d] = -2.f * r[d];
    cb_aug[t * AUGW + 8]  = n2;
    cb_aug[t * AUGW + 9]  = 0.f;
    cb_aug[t * AUGW + 10] = 0.f;
    cb_aug[t * AUGW + 11] = 0.f;
  }
  w_in_pad[t] = (t < DCODE * IN_DIM) ? w_in[t] : 0.f;    // t spans 16*512
  if (blockIdx.x == 0 && threadIdx.x < 2) loss_slots[threadIdx.x] = 0.f;
}

// ---------------- kernel 1: in_proj  z_e = z @ w_in^T + b_in ----------------
// One wave owns TWO 16-token tiles (32 tokens) so each B fragment feeds 2 WMMAs.
// Fragment loads are batched (24 loads -> 16 WMMAs) for memory-level parallelism.
__global__ __launch_bounds__(256)
void vq_inproj(const float* __restrict__ z,
               const float* __restrict__ w_in_pad,
               const float* __restrict__ b_in,
               float* __restrict__ ze) {
  const int lane = threadIdx.x & 31;
  const int wave = (blockIdx.x * blockDim.x + threadIdx.x) >> 5;
  const int tok0 = wave * 32;
  const int m    = lane & 15;     // A row (token within tile) / B column (out dim)
  const int hi   = lane >> 4;     // 0 -> K pair {0,1}, 1 -> K pair {2,3}

  const float* zrow0 = z + (size_t)(tok0 + m) * IN_DIM;
  const float* zrow1 = z + (size_t)(tok0 + 16 + m) * IN_DIM;
  const float* wrow  = w_in_pad + (size_t)m * IN_DIM;   // rows 8..15 are zero

  v8f acc0 = {}, acc1 = {};
  for (int k0 = 0; k0 < IN_DIM; k0 += 32) {             // 8 k-steps per batch
    __builtin_prefetch(zrow0 + k0 + 128, 0, 0);         // global_prefetch_b8
    __builtin_prefetch(zrow1 + k0 + 128, 0, 0);
    v2f a0[8], a1[8], b[8];
#pragma unroll
    for (int j = 0; j < 8; ++j) {
      const int kk = k0 + 4 * j + 2 * hi;
      a0[j] = *(const v2f*)(zrow0 + kk);
      a1[j] = *(const v2f*)(zrow1 + kk);
      b[j]  = *(const v2f*)(wrow + kk);
    }
#pragma unroll
    for (int j = 0; j < 8; ++j) {
      acc0 = __builtin_amdgcn_wmma_f32_16x16x4_f32(false, a0[j], false, b[j], (short)0, acc0, false, false);
      acc1 = __builtin_amdgcn_wmma_f32_16x16x4_f32(false, a1[j], false, b[j], (short)0, acc1, false, false);
    }
  }

  const float bias = (m < DCODE) ? b_in[m] : 0.f;
#pragma unroll
  for (int r = 0; r < 8; ++r) { acc0[r] += bias; acc1[r] += bias; }

  if (m < DCODE) {  // store valid columns: row M = r + 8*hi, col = m
#pragma unroll
    for (int r = 0; r < 8; ++r) {
      ze[(size_t)(tok0 + r + 8 * hi) * DCODE + m]      = acc0[r];
      ze[(size_t)(tok0 + 16 + r + 8 * hi) * DCODE + m] = acc1[r];
    }
  }
}

// ------- kernel 2: distances+argmin (WMMA), z_q gather, losses, out_proj -------
__global__ __launch_bounds__(256)
void vq_main(const float* __restrict__ ze,
             const float* __restrict__ cb_aug,
             const float* __restrict__ codebook,
             const float* __restrict__ w_out,
             const float* __restrict__ b_out,
             float* __restrict__ out_zq,
             float* __restrict__ out_idx,
             float* __restrict__ out_loss) {
  __shared__ float zq_s[8][16 * 9];  // per-wave z_q tile, stride 9 vs 64 banks

  const int lane  = threadIdx.x & 31;
  const int wslot = threadIdx.x >> 5;
  const int wave  = (blockIdx.x * blockDim.x + threadIdx.x) >> 5;
  const int tok0  = wave * 16;
  const int m     = lane & 15;
  const int hi    = lane >> 4;
  const int roff  = 8 * hi;

  // B fragments = augmented z_e^T (K=0..11, N=token). Chunk2 is the constant
  // [1,0,0,0] augmentation multiplying the ||c||^2 column of cb_aug.
  v2f b0, b1, b2;
  {
    const float* zr = ze + (size_t)(tok0 + m) * DCODE + 2 * hi;
    b0 = *(const v2f*)(zr + 0);
    b1 = *(const v2f*)(zr + 4);
    b2 = (v2f){hi ? 0.f : 1.f, 0.f};  // K=8 -> 1.0 (lo half), K=9..11 -> 0
  }

  // score(code, tok) = ||c||^2 - 2 c.z_e  (||z_e||^2 constant per token).
  // Two 16-code tiles per iteration: 6 batched loads -> 6 WMMAs.
  float minv = 3.4e38f;
  int   mini = 0;
  for (int c0 = 0; c0 < KCODES; c0 += 32) {
    const float* arA = cb_aug + (size_t)(c0 + m) * AUGW + 2 * hi;
    const float* arB = cb_aug + (size_t)(c0 + 16 + m) * AUGW + 2 * hi;
    v2f a0 = *(const v2f*)(arA + 0);
    v2f a1 = *(const v2f*)(arA + 4);
    v2f a2 = *(const v2f*)(arA + 8);
    v2f a3 = *(const v2f*)(arB + 0);
    v2f a4 = *(const v2f*)(arB + 4);
    v2f a5 = *(const v2f*)(arB + 8);
    v8f accA = {}, accB = {};
    accA = __builtin_amdgcn_wmma_f32_16x16x4_f32(false, a0, false, b0, (short)0, accA, false, false);
    accB = __builtin_amdgcn_wmma_f32_16x16x4_f32(false, a3, false, b0, (short)0, accB, false, false);
    accA = __builtin_amdgcn_wmma_f32_16x16x4_f32(false, a1, false, b1, (short)0, accA, false, false);
    accB = __builtin_amdgcn_wmma_f32_16x16x4_f32(false, a4, false, b1, (short)0, accB, false, false);
    accA = __builtin_amdgcn_wmma_f32_16x16x4_f32(false, a2, false, b2, (short)0, accA, false, false);
    accB = __builtin_amdgcn_wmma_f32_16x16x4_f32(false, a5, false, b2, (short)0, accB, false, false);
#pragma unroll
    for (int r = 0; r < 8; ++r) {          // lane's codes: M = r + roff
      const float sA = accA[r];
      const int   cA = c0 + roff + r;
      if (sA < minv) { minv = sA; mini = cA; }
      const float sB = accB[r];
      const int   cB = c0 + 16 + roff + r;
      if (sB < minv) { minv = sB; mini = cB; }
    }
  }

  // Combine the two half-wave candidates per token (tie -> lower index).
  const float mv_o = __shfl(minv, lane + 16, 32);
  const int   mi_o = __shfl(mini, lane + 16, 32);
  float dsq = 0.f;
  if (lane < 16) {
    if (mv_o < minv || (mv_o == minv && mi_o < mini)) { minv = mv_o; mini = mi_o; }
    out_idx[tok0 + lane] = (float)mini;  // argmin index, float-encoded
    const float* zr = ze + (size_t)(tok0 + lane) * DCODE;
    const float* cr = codebook + (size_t)mini * DCODE;
#pragma unroll
    for (int d = 0; d < DCODE; ++d) {
      const float c = cr[d];
      const float diff = zr[d] - c;
      dsq += diff * diff;
      zq_s[wslot][lane * 9 + d] = c;      // z_q tile for out_proj
    }
  }
  __syncthreads();

  // losses: commitment == codebook loss in forward value
#pragma unroll
  for (int off = 16; off >= 1; off >>= 1) dsq += __shfl_down(dsq, off, 32);
  if (lane == 0) {
    const float inv = 1.0f / (float)((size_t)NTOK * DCODE);
    atomicAdd(&out_loss[0], dsq * inv);
    atomicAdd(&out_loss[1], dsq * inv);
  }

  // out_proj: [16 tok x 8] @ [8 x 512] via WMMA (forward z_q_st == z_q).
  // Two 16-output tiles per iteration.
  v2f a0 = (v2f){zq_s[wslot][m * 9 + 2 * hi + 0], zq_s[wslot][m * 9 + 2 * hi + 1]};
  v2f a1 = (v2f){zq_s[wslot][m * 9 + 4 + 2 * hi + 0], zq_s[wslot][m * 9 + 4 + 2 * hi + 1]};
  for (int n0 = 0; n0 < IN_DIM; n0 += 32) {
    const int nA = n0 + m;
    const int nB = n0 + 16 + m;
    const float* wrA = w_out + (size_t)nA * DCODE + 2 * hi;
    const float* wrB = w_out + (size_t)nB * DCODE + 2 * hi;
    v2f wA0 = *(const v2f*)(wrA + 0);
    v2f wA1 = *(const v2f*)(wrA + 4);
    v2f wB0 = *(const v2f*)(wrB + 0);
    v2f wB1 = *(const v2f*)(wrB + 4);
    v8f accA = {}, accB = {};
    accA = __builtin_amdgcn_wmma_f32_16x16x4_f32(false, a0, false, wA0, (short)0, accA, false, false);
    accB = __builtin_amdgcn_wmma_f32_16x16x4_f32(false, a0, false, wB0, (short)0, accB, false, false);
    accA = __builtin_amdgcn_wmma_f32_16x16x4_f32(false, a1, false, wA1, (short)0, accA, false, false);
    accB = __builtin_amdgcn_wmma_f32_16x16x4_f32(false, a1, false, wB1, (short)0, accB, false, false);
    const float biasA = b_out[nA];
    const float biasB = b_out[nB];
#pragma unroll
    for (int r = 0; r < 8; ++r) {
      const size_t row = (size_t)(tok0 + r + roff) * IN_DIM;
      out_zq[row + nA] = accA[r] + biasA;
      out_zq[row + nB] = accB[r] + biasB;
    }
  }
}

extern "C" void kernel_launch(void* const* d_in, const int* in_sizes, int n_in,
                              void* d_out, int out_size, void* d_ws, size_t ws_size,
                              hipStream_t stream) {
  const float* z        = (const float*)d_in[0];
  const float* w_in     = (const float*)d_in[1];
  const float* b_in     = (const float*)d_in[2];
  const float* w_out    = (const float*)d_in[3];
  const float* b_out    = (const float*)d_in[4];
  const float* codebook = (const float*)d_in[5];

  float* out      = (float*)d_out;
  float* out_zq   = out;                                  // [16,4096,512]
  float* out_idx  = out + (size_t)NTOK * IN_DIM;          // [16,4096]
  float* out_loss = out_idx + NTOK;                       // 2 scalars

  // workspace: z_e (2 MB) + augmented codebook (48 KB) + padded w_in (32 KB)
  float* ze       = (float*)d_ws;
  float* cb_aug   = ze + (size_t)NTOK * DCODE;
  float* w_in_pad = cb_aug + (size_t)KCODES * AUGW;

  vq_prep  <<<32, 256, 0, stream>>>(codebook, w_in, cb_aug, w_in_pad, out_loss);
  vq_inproj<<<NTOK / 32 / 8, 256, 0, stream>>>(z, w_in_pad, b_in, ze);
  vq_main  <<<NTOK / 16 / 8, 256, 0, stream>>>(ze, cb_aug, codebook, w_out,
                                               b_out, out_zq, out_idx, out_loss);
}